// LrState_7017976562041
// MI455X (gfx1250) — compile-verified
//
#include <hip/hip_runtime.h>
#include <hip/hip_bf16.h>

typedef __attribute__((ext_vector_type(2))) float v2f;
typedef __attribute__((ext_vector_type(8))) float v8f;

#define SEQ    2048
#define NHEAD  16
#define DK     64
#define DMODEL 1024

// One wave computes one 16x16 tile of scores[b,h,:,:] via 16 chained
// V_WMMA_F32_16X16X4_F32 over the K=64 head dim.
__global__ __launch_bounds__(256) void lrstate_scores_kernel(
    const float* __restrict__ qry,
    const float* __restrict__ key,
    const float* __restrict__ gammas,
    const int*   __restrict__ sw,
    const int*   __restrict__ peek_ptr,
    float*       __restrict__ out)
{
    const int lane = threadIdx.x & 31;
    const int wave = threadIdx.x >> 5;

    const int q0 = blockIdx.y * 16;
    const int k0 = (blockIdx.x * 8 + wave) * 16;
    const int bh = blockIdx.z;
    const int b  = bh >> 4;
    const int h  = bh & 15;

    const int peek = __builtin_amdgcn_readfirstlane(peek_ptr[0]);

    const int lrow = lane & 15;   // A: M row / B: N col / store: k col
    const int lhi  = lane >> 4;   // lane-half selects K pair (loads) and M+8 (store)

    float* outBase = out + (((size_t)(b * NHEAD + h) * SEQ + q0) * SEQ + k0);

    // tile fully above causal diagonal (k > q-1+peek for every element) -> zeros
    const bool fullMasked = (k0 > q0 + 14 + peek);
    if (fullMasked) {
#pragma unroll
        for (int v = 0; v < 8; ++v)
            outBase[(size_t)(v + 8 * lhi) * SEQ + lrow] = 0.0f;
        return;
    }

    // scale = clamp(exp(-softplus(g)),1e-5,1e5) / sqrt(1024)
    //       = clamp(sigmoid(-g),...) * (1/32)
    float g  = gammas[h];
    float te = 1.0f / (1.0f + __expf(g));
    te = fminf(fmaxf(te, 1e-5f), 1e5f);
    const float scale = te * 0.03125f;

    // Per-lane contiguous float2 loads matching the 16x4 f32 A/B striping.
    const float* aP = qry + ((size_t)(b * SEQ + q0 + lrow) * DMODEL + h * DK + 2 * lhi);
    const float* bP = key + ((size_t)(b * SEQ + k0 + lrow) * DMODEL + h * DK + 2 * lhi);

    v2f a[16], bb[16];
#pragma unroll
    for (int t = 0; t < 16; ++t) {
        a[t]  = *(const v2f*)(aP + 4 * t);
        bb[t] = *(const v2f*)(bP + 4 * t);
    }

    v8f c = {};
#pragma unroll
    for (int t = 0; t < 16; ++t) {
        c = __builtin_amdgcn_wmma_f32_16x16x4_f32(
                /*neg_a=*/false, a[t], /*neg_b=*/false, bb[t],
                /*c_mod=*/(short)0, c, /*reuse_a=*/false, /*reuse_b=*/false);
    }

    // Apply decay scale + state/causal mask, store (64B-coalesced per half-wave).
    const int kcol = k0 + lrow;
#pragma unroll
    for (int v = 0; v < 8; ++v) {
        const int qrow = q0 + v + 8 * lhi;
        const int m    = sw[((size_t)b * SEQ + qrow) * SEQ + kcol];
        const bool keep = (m != 0) && (kcol <= qrow - 1 + peek);
        outBase[(size_t)(v + 8 * lhi) * SEQ + lrow] = keep ? c[v] * scale : 0.0f;
    }
}

extern "C" void kernel_launch(void* const* d_in, const int* in_sizes, int n_in,
                              void* d_out, int out_size, void* d_ws, size_t ws_size,
                              hipStream_t stream) {
    (void)in_sizes; (void)n_in; (void)out_size; (void)d_ws; (void)ws_size;
    const float* qry    = (const float*)d_in[0];
    const float* key    = (const float*)d_in[1];
    const float* gammas = (const float*)d_in[2];
    const int*   sw     = (const int*)d_in[3];
    const int*   peek   = (const int*)d_in[4];
    float*       out    = (float*)d_out;

    // 2048/16 = 128 tiles per dim; 8 k-tiles per block (one per wave).
    dim3 grid(128 / 8, 128, 2 * NHEAD);
    dim3 block(256);
    hipLaunchKernelGGL(lrstate_scores_kernel, grid, block, 0, stream,
                       qry, key, gammas, sw, peek, out);
}